// VariationalGCNEncoder_54597624267033
// MI455X (gfx1250) — compile-verified
//
#include <hip/hip_runtime.h>
#include <hip/hip_bf16.h>

typedef __attribute__((ext_vector_type(2))) float v2f;
typedef __attribute__((ext_vector_type(8))) float v8f;

#define NN   100000
#define NE   1600000
#define CIN  128
#define CH   48
#define CO   32

// ---------------- init: deg=1 (self-loop), zero h-accum and d_out (vectorized) ----------------
__global__ __launch_bounds__(256) void k_init(float* __restrict__ deg,
                                              float4* __restrict__ h4,
                                              float4* __restrict__ out4) {
  long long stride = (long long)gridDim.x * blockDim.x;
  long long total  = (long long)NN * 16;           // d_out = NN*64 floats = NN*16 float4
  const float4 z = make_float4(0.f, 0.f, 0.f, 0.f);
  for (long long i = (long long)blockIdx.x * blockDim.x + threadIdx.x; i < total; i += stride) {
    out4[i] = z;
    if (i < (long long)NN * 12) h4[i] = z;         // h = NN*48 floats
    if (i < NN) deg[i] = 1.0f;
  }
}

// ---------------- degree accumulation over real edges ----------------
__global__ __launch_bounds__(256) void k_deg(const int* __restrict__ dst,
                                             float* __restrict__ deg) {
  long long stride = (long long)gridDim.x * blockDim.x;
  for (long long e = (long long)blockIdx.x * blockDim.x + threadIdx.x; e < NE; e += stride)
    atomicAdd(&deg[dst[e]], 1.0f);
}

// ---------------- deg -> dinv (in place) ----------------
__global__ __launch_bounds__(256) void k_dinv(float* __restrict__ deg) {
  int i = blockIdx.x * blockDim.x + threadIdx.x;
  if (i < NN) deg[i] = __frsqrt_rn(deg[i]);        // deg >= 1 always (self-loop)
}

// ---------------- GEMM1: t1 = x @ W1   (100000x128 @ 128x48), fp32 WMMA ----------------
__global__ __launch_bounds__(256) void k_gemm1(const float* __restrict__ x,
                                               const float* __restrict__ W,
                                               float* __restrict__ t1) {
  int wave = (blockIdx.x * blockDim.x + threadIdx.x) >> 5;
  int lane = threadIdx.x & 31;
  if (wave >= NN / 16) return;               // wave-uniform guard: EXEC all-1s inside

  int row = wave * 16 + (lane & 15);         // A: lane%16 = M row
  int kh  = (lane >> 4) << 1;                // A/B: lane/16 selects K pair {0,1} vs {2,3}
  int col = lane & 15;                       // B: lane%16 = N col

  v8f acc0 = {}, acc1 = {}, acc2 = {};
  const float* xr = x + (long long)row * CIN;

  for (int kk = 0; kk < CIN; kk += 4) {
    v2f a = { xr[kk + kh], xr[kk + kh + 1] };
    const float* w0 = W + (kk + kh) * CH;
    const float* w1 = W + (kk + kh + 1) * CH;
    v2f b0 = { w0[col],      w1[col]      };
    v2f b1 = { w0[col + 16], w1[col + 16] };
    v2f b2 = { w0[col + 32], w1[col + 32] };
    acc0 = __builtin_amdgcn_wmma_f32_16x16x4_f32(false, a, false, b0, (short)0, acc0, false, false);
    acc1 = __builtin_amdgcn_wmma_f32_16x16x4_f32(false, a, false, b1, (short)0, acc1, false, false);
    acc2 = __builtin_amdgcn_wmma_f32_16x16x4_f32(false, a, false, b2, (short)0, acc2, false, false);
  }

  int mrow = wave * 16 + (lane >> 4) * 8;    // C/D: M = j + 8*(lane/16)
  int ncol = lane & 15;
  for (int j = 0; j < 8; ++j) {
    long long base = (long long)(mrow + j) * CH + ncol;
    t1[base]      = acc0[j];
    t1[base + 16] = acc1[j];
    t1[base + 32] = acc2[j];
  }
}

// ---------------- scatter layer 1: h[dst] += dinv[s]*dinv[d] * t1[src]  (48 ch, x4 vec) ----------------
__global__ __launch_bounds__(256) void k_scatter48(const int* __restrict__ src,
                                                   const int* __restrict__ dst,
                                                   const float* __restrict__ dinv,
                                                   const float* __restrict__ t1,
                                                   float* __restrict__ h) {
  long long stride = (long long)gridDim.x * blockDim.x;
  long long total  = (long long)NE * 12;           // 12 float4 groups per edge
  for (long long i = (long long)blockIdx.x * blockDim.x + threadIdx.x; i < total; i += stride) {
    int e = (int)(i / 12);
    int q = (int)(i - (long long)e * 12);
    int s = src[e], d = dst[e];
    float nrm = dinv[s] * dinv[d];
    const float4 v = *(const float4*)(t1 + (long long)s * CH + q * 4);
    float* hp = h + (long long)d * CH + q * 4;
    atomicAdd(hp + 0, nrm * v.x);
    atomicAdd(hp + 1, nrm * v.y);
    atomicAdd(hp + 2, nrm * v.z);
    atomicAdd(hp + 3, nrm * v.w);
  }
}

// ---------------- finalize layer 1: self-loop + bias + ReLU (in place on h, x4 vec) ----------------
__global__ __launch_bounds__(256) void k_fin1(const float* __restrict__ dinv,
                                              const float* __restrict__ t1,
                                              const float* __restrict__ b1,
                                              float* __restrict__ h) {
  long long stride = (long long)gridDim.x * blockDim.x;
  long long total  = (long long)NN * 12;
  for (long long i = (long long)blockIdx.x * blockDim.x + threadIdx.x; i < total; i += stride) {
    int node = (int)(i / 12);
    int q    = (int)(i - (long long)node * 12);
    float w  = dinv[node] * dinv[node];
    float4 hv = *(const float4*)(h  + (long long)node * CH + q * 4);
    float4 tv = *(const float4*)(t1 + (long long)node * CH + q * 4);
    float4 bv = *(const float4*)(b1 + q * 4);
    float4 r;
    r.x = fmaxf(hv.x + w * tv.x + bv.x, 0.f);
    r.y = fmaxf(hv.y + w * tv.y + bv.y, 0.f);
    r.z = fmaxf(hv.z + w * tv.z + bv.z, 0.f);
    r.w = fmaxf(hv.w + w * tv.w + bv.w, 0.f);
    *(float4*)(h + (long long)node * CH + q * 4) = r;
  }
}

// ---------------- GEMM2 fused: tmu = h@Wmu, tls = h@Wls  (100000x48 @ 48x32) ----------------
__global__ __launch_bounds__(256) void k_gemm2(const float* __restrict__ h,
                                               const float* __restrict__ Wmu,
                                               const float* __restrict__ Wls,
                                               float* __restrict__ tmu,
                                               float* __restrict__ tls) {
  int wave = (blockIdx.x * blockDim.x + threadIdx.x) >> 5;
  int lane = threadIdx.x & 31;
  if (wave >= NN / 16) return;

  int row = wave * 16 + (lane & 15);
  int kh  = (lane >> 4) << 1;
  int col = lane & 15;

  v8f am0 = {}, am1 = {}, al0 = {}, al1 = {};
  const float* hr = h + (long long)row * CH;

  for (int kk = 0; kk < CH; kk += 4) {
    v2f a = { hr[kk + kh], hr[kk + kh + 1] };
    const float* m0 = Wmu + (kk + kh) * CO;
    const float* m1 = Wmu + (kk + kh + 1) * CO;
    const float* l0 = Wls + (kk + kh) * CO;
    const float* l1 = Wls + (kk + kh + 1) * CO;
    v2f bm0 = { m0[col],      m1[col]      };
    v2f bm1 = { m0[col + 16], m1[col + 16] };
    v2f bl0 = { l0[col],      l1[col]      };
    v2f bl1 = { l0[col + 16], l1[col + 16] };
    am0 = __builtin_amdgcn_wmma_f32_16x16x4_f32(false, a, false, bm0, (short)0, am0, false, false);
    am1 = __builtin_amdgcn_wmma_f32_16x16x4_f32(false, a, false, bm1, (short)0, am1, false, false);
    al0 = __builtin_amdgcn_wmma_f32_16x16x4_f32(false, a, false, bl0, (short)0, al0, false, false);
    al1 = __builtin_amdgcn_wmma_f32_16x16x4_f32(false, a, false, bl1, (short)0, al1, false, false);
  }

  int mrow = wave * 16 + (lane >> 4) * 8;
  int ncol = lane & 15;
  for (int j = 0; j < 8; ++j) {
    long long base = (long long)(mrow + j) * CO + ncol;
    tmu[base]      = am0[j];
    tmu[base + 16] = am1[j];
    tls[base]      = al0[j];
    tls[base + 16] = al1[j];
  }
}

// ---------------- scatter layer 2: mu & logstd accumulate into d_out (x4 vec) ----------------
__global__ __launch_bounds__(256) void k_scatter64(const int* __restrict__ src,
                                                   const int* __restrict__ dst,
                                                   const float* __restrict__ dinv,
                                                   const float* __restrict__ tmu,
                                                   const float* __restrict__ tls,
                                                   float* __restrict__ out) {
  long long stride = (long long)gridDim.x * blockDim.x;
  long long total  = (long long)NE * 16;           // 8 mu groups + 8 ls groups per edge
  for (long long i = (long long)blockIdx.x * blockDim.x + threadIdx.x; i < total; i += stride) {
    int e = (int)(i >> 4);
    int q = (int)(i & 15);
    int s = src[e], d = dst[e];
    float nrm = dinv[s] * dinv[d];
    const float* tsrc;
    float* odst;
    if (q < 8) {
      tsrc = tmu + (long long)s * CO + q * 4;
      odst = out + (long long)d * CO + q * 4;
    } else {
      int g = q - 8;
      tsrc = tls + (long long)s * CO + g * 4;
      odst = out + (long long)NN * CO + (long long)d * CO + g * 4;
    }
    const float4 v = *(const float4*)tsrc;
    atomicAdd(odst + 0, nrm * v.x);
    atomicAdd(odst + 1, nrm * v.y);
    atomicAdd(odst + 2, nrm * v.z);
    atomicAdd(odst + 3, nrm * v.w);
  }
}

// ---------------- finalize layer 2: self-loop + bias (x4 vec) ----------------
__global__ __launch_bounds__(256) void k_fin2(const float* __restrict__ dinv,
                                              const float4* __restrict__ tmu4,
                                              const float4* __restrict__ tls4,
                                              const float4* __restrict__ bmu4,
                                              const float4* __restrict__ bls4,
                                              float4* __restrict__ out4) {
  long long stride = (long long)gridDim.x * blockDim.x;
  long long total  = (long long)NN * 8;            // NN*32 floats per half = NN*8 float4
  for (long long i = (long long)blockIdx.x * blockDim.x + threadIdx.x; i < total; i += stride) {
    int node = (int)(i >> 3);
    int q    = (int)(i & 7);
    float w  = dinv[node] * dinv[node];
    float4 m = out4[i], tm = tmu4[i], bm = bmu4[q];
    m.x += w * tm.x + bm.x;  m.y += w * tm.y + bm.y;
    m.z += w * tm.z + bm.z;  m.w += w * tm.w + bm.w;
    out4[i] = m;
    float4 l = out4[(long long)NN * 8 + i], tl = tls4[i], bl = bls4[q];
    l.x += w * tl.x + bl.x;  l.y += w * tl.y + bl.y;
    l.z += w * tl.z + bl.z;  l.w += w * tl.w + bl.w;
    out4[(long long)NN * 8 + i] = l;
  }
}

extern "C" void kernel_launch(void* const* d_in, const int* in_sizes, int n_in,
                              void* d_out, int out_size, void* d_ws, size_t ws_size,
                              hipStream_t stream) {
  const float* x   = (const float*)d_in[0];
  const int*   ei  = (const int*)d_in[1];   // [2, NE] flat: row0=src, row1=dst
  const float* W1  = (const float*)d_in[2];
  const float* b1  = (const float*)d_in[3];
  const float* Wmu = (const float*)d_in[4];
  const float* bmu = (const float*)d_in[5];
  const float* Wls = (const float*)d_in[6];
  const float* bls = (const float*)d_in[7];
  float* out = (float*)d_out;

  const int* src = ei;
  const int* dst = ei + NE;

  // workspace layout (floats): deg/dinv[NN] | t1[NN*48] | h[NN*48] | tmu[NN*32] | tls[NN*32]
  // every sub-array size is a multiple of 4 floats -> all 16B-aligned for float4
  float* deg = (float*)d_ws;
  float* t1  = deg + NN;
  float* h   = t1 + (size_t)NN * CH;
  float* tmu = h  + (size_t)NN * CH;
  float* tls = tmu + (size_t)NN * CO;

  dim3 blk(256);
  int gemm_blocks = (NN / 16 + 7) / 8;   // 8 waves per 256-thread block

  k_init     <<<2048, blk, 0, stream>>>(deg, (float4*)h, (float4*)out);
  k_deg      <<<2048, blk, 0, stream>>>(dst, deg);
  k_dinv     <<<(NN + 255) / 256, blk, 0, stream>>>(deg);
  k_gemm1    <<<gemm_blocks, blk, 0, stream>>>(x, W1, t1);
  k_scatter48<<<8192, blk, 0, stream>>>(src, dst, deg, t1, h);
  k_fin1     <<<2048, blk, 0, stream>>>(deg, t1, b1, h);
  k_gemm2    <<<gemm_blocks, blk, 0, stream>>>(h, Wmu, Wls, tmu, tls);
  k_scatter64<<<8192, blk, 0, stream>>>(src, dst, deg, tmu, tls, out);
  k_fin2     <<<2048, blk, 0, stream>>>(deg, (const float4*)tmu, (const float4*)tls,
                                        (const float4*)bmu, (const float4*)bls, (float4*)out);
}